// MMDLoss_33414845563209
// MI455X (gfx1250) — compile-verified
//
#include <hip/hip_runtime.h>
#include <cstdint>

typedef __attribute__((ext_vector_type(2))) float v2f;
typedef __attribute__((ext_vector_type(8))) float v8f;

#define NPTS      16384
#define HALF_PTS  8192
#define D_DIM     256
#define TILE      128
#define KC        32
#define LSTR      34      // LDS row stride in floats (even -> 8B aligned, odd/2 -> conflict-free)
#define NTILES    128     // 16384 / 128

// -------- kernel 0: per-point squared norms (one wave per point) --------
__global__ __launch_bounds__(256) void mmd_norm_kernel(const float* __restrict__ s,
                                                       const float* __restrict__ t,
                                                       float* __restrict__ nrm) {
  const int wv   = threadIdx.x >> 5;
  const int lane = threadIdx.x & 31;
  const int p    = blockIdx.x * 8 + wv;
  const float* src = (p < HALF_PTS) ? (s + (size_t)p * D_DIM)
                                    : (t + (size_t)(p - HALF_PTS) * D_DIM);
  float a = 0.f;
#pragma unroll
  for (int k = lane; k < D_DIM; k += 32) { float v = src[k]; a += v * v; }
#pragma unroll
  for (int off = 16; off > 0; off >>= 1) a += __shfl_xor(a, off, 32);
  if (lane == 0) nrm[p] = a;
}

// -------- kernel 1: fused WMMA Gram tile + multi-bandwidth Gaussian sum --------
__global__ __launch_bounds__(256) void mmd_tile_kernel(const float* __restrict__ s,
                                                       const float* __restrict__ t,
                                                       const float* __restrict__ nrm,
                                                       float* __restrict__ partial) {
  __shared__ float sA[TILE * LSTR];
  __shared__ float sB[TILE * LSTR];
  __shared__ float nA[TILE];
  __shared__ float nB[TILE];
  __shared__ float wsum[8];

  const int bj   = blockIdx.x;
  const int bi   = blockIdx.y;
  const int pidx = bi * NTILES + bj;
  if (bi > bj) {                       // lower triangle: no work, keep partial deterministic
    if (threadIdx.x == 0) partial[pidx] = 0.f;
    return;
  }

  // whole 128-row panel comes from one side (8192 % 128 == 0)
  const float* Abase = (bi < 64) ? (s + (size_t)bi * TILE * D_DIM)
                                 : (t + (size_t)(bi - 64) * TILE * D_DIM);
  const float* Bbase = (bj < 64) ? (s + (size_t)bj * TILE * D_DIM)
                                 : (t + (size_t)(bj - 64) * TILE * D_DIM);

  const int tid = threadIdx.x;
  if (tid < TILE) nA[tid]        = nrm[bi * TILE + tid];
  else            nB[tid - TILE] = nrm[bj * TILE + (tid - TILE)];

  const int wv    = tid >> 5;
  const int lane  = tid & 31;
  const int wm    = wv >> 2;           // 2 wave-rows
  const int wn    = wv & 3;            // 4 wave-cols
  const int rbase = wm * 64;           // 64x32 sub-tile per wave -> 4x2 wmma frags
  const int cbase = wn * 32;
  const int lrow  = lane & 15;
  const int khalf = (lane >> 4) * 2;   // upper lane-half owns K=2,3 (f32 16x16x4 layout)

  v8f c[4][2];
  v8f zero = {};
#pragma unroll
  for (int fm = 0; fm < 4; ++fm)
#pragma unroll
    for (int fn = 0; fn < 2; ++fn) c[fm][fn] = zero;

  for (int kc = 0; kc < D_DIM; kc += KC) {
    // stage 128x32 A and B panels into LDS (float4 global loads, 8B LDS stores)
#pragma unroll
    for (int ci = 0; ci < 4; ++ci) {
      int f4  = tid + ci * 256;        // 1024 float4 per panel
      int row = f4 >> 3;               // 8 float4 per 32-float row
      int col = (f4 & 7) * 4;
      float4 va = *reinterpret_cast<const float4*>(Abase + row * D_DIM + kc + col);
      float4 vb = *reinterpret_cast<const float4*>(Bbase + row * D_DIM + kc + col);
      float* da = &sA[row * LSTR + col];
      da[0] = va.x; da[1] = va.y; da[2] = va.z; da[3] = va.w;
      float* db = &sB[row * LSTR + col];
      db[0] = vb.x; db[1] = vb.y; db[2] = vb.z; db[3] = vb.w;
    }
    __syncthreads();

#pragma unroll
    for (int k = 0; k < KC; k += 4) {
      v2f a[4];
      v2f b[2];
#pragma unroll
      for (int fm = 0; fm < 4; ++fm)
        a[fm] = *reinterpret_cast<const v2f*>(
            &sA[(rbase + fm * 16 + lrow) * LSTR + k + khalf]);
#pragma unroll
      for (int fn = 0; fn < 2; ++fn)
        b[fn] = *reinterpret_cast<const v2f*>(
            &sB[(cbase + fn * 16 + lrow) * LSTR + k + khalf]);
#pragma unroll
      for (int fm = 0; fm < 4; ++fm)
#pragma unroll
        for (int fn = 0; fn < 2; ++fn)
          c[fm][fn] = __builtin_amdgcn_wmma_f32_16x16x4_f32(
              false, a[fm], false, b[fn], (short)0, c[fm][fn], false, false);
    }
    __syncthreads();
  }

  // epilogue: dist^2 -> sum of 5 Gaussian kernels (exp2 with folded constants)
  constexpr float LOG2E = 1.4426950408889634f;
  const float k0 = -LOG2E / (2.f * 0.2f * 0.2f);
  const float k1 = -LOG2E / (2.f * 0.5f * 0.5f);
  const float k2 = -LOG2E / (2.f * 1.0f * 1.0f);
  const float k3 = -LOG2E / (2.f * 2.0f * 2.0f);
  const float k4 = -LOG2E / (2.f * 5.0f * 5.0f);

  float acc = 0.f;
  const int mhi = (lane >> 4) * 8;     // C layout: upper lanes hold M = e + 8
#pragma unroll
  for (int fm = 0; fm < 4; ++fm) {
#pragma unroll
    for (int fn = 0; fn < 2; ++fn) {
      float yn = nB[cbase + fn * 16 + lrow];
#pragma unroll
      for (int e = 0; e < 8; ++e) {
        float xn = nA[rbase + fm * 16 + mhi + e];
        float d  = xn + yn - 2.f * c[fm][fn][e];
        d = fmaxf(d, 0.f);
        acc += __builtin_amdgcn_exp2f(d * k0);
        acc += __builtin_amdgcn_exp2f(d * k1);
        acc += __builtin_amdgcn_exp2f(d * k2);
        acc += __builtin_amdgcn_exp2f(d * k3);
        acc += __builtin_amdgcn_exp2f(d * k4);
      }
    }
  }

#pragma unroll
  for (int off = 16; off > 0; off >>= 1) acc += __shfl_xor(acc, off, 32);
  if (lane == 0) wsum[wv] = acc;
  __syncthreads();
  if (tid == 0) {
    float sgn = ((bi < 64) == (bj < 64)) ? 1.f : -1.f;   // same-side vs cross
    float wgt = (bi == bj) ? 1.f : 2.f;                  // symmetry weight
    float ssum = 0.f;
#pragma unroll
    for (int i = 0; i < 8; ++i) ssum += wsum[i];
    partial[pidx] = ssum * sgn * wgt;
  }
}

// -------- kernel 2: deterministic final reduction --------
__global__ __launch_bounds__(256) void mmd_reduce_kernel(const float* __restrict__ partial,
                                                         float* __restrict__ out) {
  __shared__ float sm[256];
  float a = 0.f;
  for (int i = threadIdx.x; i < NTILES * NTILES; i += 256) a += partial[i];
  sm[threadIdx.x] = a;
  __syncthreads();
  for (int s2 = 128; s2 > 0; s2 >>= 1) {
    if (threadIdx.x < s2) sm[threadIdx.x] += sm[threadIdx.x + s2];
    __syncthreads();
  }
  if (threadIdx.x == 0)
    out[0] = sm[0] * (1.0f / (5.0f * 8192.0f * 8192.0f));
}

extern "C" void kernel_launch(void* const* d_in, const int* in_sizes, int n_in,
                              void* d_out, int out_size, void* d_ws, size_t ws_size,
                              hipStream_t stream) {
  const float* s = (const float*)d_in[0];   // s_feats [8192,256] f32
  const float* t = (const float*)d_in[1];   // t_feats [8192,256] f32
  float* nrm     = (float*)d_ws;            // 16384 floats
  float* partial = nrm + NPTS;              // 16384 floats
  float* out     = (float*)d_out;           // scalar f32

  mmd_norm_kernel<<<NPTS / 8, 256, 0, stream>>>(s, t, nrm);
  mmd_tile_kernel<<<dim3(NTILES, NTILES), 256, 0, stream>>>(s, t, nrm, partial);
  mmd_reduce_kernel<<<1, 256, 0, stream>>>(partial, out);
}